// GCN_17703855194320
// MI455X (gfx1250) — compile-verified
//
#include <hip/hip_runtime.h>
#include <hip/hip_bf16.h>

// ---------------------------------------------------------------------------
// 2-layer GCN for MI455X (gfx1250, wave32).
//   layer: h = X @ W ;  out = h * dinv^2 + b ;  out[dst] += dinv[s]*dinv[d] * h[src]
// GEMMs use V_WMMA_F32_16X16X4_F32. W is pre-padded to stride 320 with zero
// tail columns so the WMMA inner loop has NO conditional loads (branch-free).
// Scatter uses float4 gathers + global_atomic_add_f32 (L2-resident buffers).
// ---------------------------------------------------------------------------

#define N_NODES 50000
#define N_EDGES 800000
#define N_FEAT  128
#define HIDDEN  300
#define NPAD    320     // HIDDEN padded to 5 x 64-wide column groups

typedef __attribute__((ext_vector_type(2))) float v2f;
typedef __attribute__((ext_vector_type(8))) float v8f;

// ---------------- degree / norm precompute ----------------

__global__ void gcn_deg_init(float* __restrict__ deg, int n) {
    int i = blockIdx.x * blockDim.x + threadIdx.x;
    if (i < n) deg[i] = 1.0f;               // self loop
}

__global__ void gcn_deg_accum(float* __restrict__ deg, const int* __restrict__ dst, int e) {
    int i = blockIdx.x * blockDim.x + threadIdx.x;
    if (i < e) atomicAdd(&deg[dst[i]], 1.0f);
}

__global__ void gcn_rsqrt(float* __restrict__ deg, int n) {
    int i = blockIdx.x * blockDim.x + threadIdx.x;
    if (i < n) deg[i] = rsqrtf(deg[i]);     // deg -> dinv in place
}

__global__ void gcn_norm(const float* __restrict__ dinv, const int* __restrict__ src,
                         const int* __restrict__ dst, float* __restrict__ norm, int e) {
    int i = blockIdx.x * blockDim.x + threadIdx.x;
    if (i < e) norm[i] = dinv[src[i]] * dinv[dst[i]];
}

// ---------------- pad W[K,HIDDEN] -> Wp[K,NPAD] (zero tail) ----------------

__global__ void gcn_pad_w(const float* __restrict__ W, float* __restrict__ Wp, int K) {
    int i = blockIdx.x * blockDim.x + threadIdx.x;
    if (i >= K * NPAD) return;
    int k = i / NPAD;
    int n = i - k * NPAD;
    Wp[i] = (n < HIDDEN) ? W[k * HIDDEN + n] : 0.0f;
}

// ---------------- fp32 WMMA GEMM: C[M,N] = A[M,K] * Wp[K,NPAD] ----------------
// One wave computes a 16x64 tile: 4 accumulators, 4 WMMAs per K-step of 4.
// All inner-loop loads are unconditional (Wp is padded); only the C-store
// epilogue is column-guarded. M % 16 == 0 required.

__global__ void gcn_gemm_wmma(const float* __restrict__ A, const float* __restrict__ Wp,
                              float* __restrict__ C, int M, int K, int N, int colGroups) {
    int wave = (blockIdx.x * blockDim.x + threadIdx.x) >> 5;
    int lane = threadIdx.x & 31;
    int totalWaves = (M >> 4) * colGroups;
    if (wave >= totalWaves) return;         // wave-uniform: EXEC stays all-ones

    int tm = wave / colGroups;
    int tn = wave - tm * colGroups;
    int m0 = tm << 4;
    int n0 = tn << 6;

    int lrow  = lane & 15;                  // 0..15
    int khalf = lane >> 4;                  // 0/1 -> K offset 0/2 (ISA A/B layout)

    const float* Arow = A + (size_t)(m0 + lrow) * K + 2 * khalf;
    // B base for this lane: row (2*khalf) of the padded W, at this lane's column
    const float* Wl = Wp + (size_t)(2 * khalf) * NPAD + n0 + lrow;

    v8f acc0 = {}, acc1 = {}, acc2 = {}, acc3 = {};

    for (int kk = 0; kk < K; kk += 4) {
        // A operand: lanes 0-15 rows m0+l with {kk,kk+1}; lanes 16-31 with {kk+2,kk+3}
        v2f a;
        a.x = Arow[kk];
        a.y = Arow[kk + 1];

        // B operand: v0 = Wp[kk+2*khalf][col], v1 = Wp[kk+1+2*khalf][col]
        const float* Wr0 = Wl + (size_t)kk * NPAD;
        const float* Wr1 = Wr0 + NPAD;

        v2f b0, b1, b2, b3;
        b0.x = Wr0[0];   b0.y = Wr1[0];
        b1.x = Wr0[16];  b1.y = Wr1[16];
        b2.x = Wr0[32];  b2.y = Wr1[32];
        b3.x = Wr0[48];  b3.y = Wr1[48];

        acc0 = __builtin_amdgcn_wmma_f32_16x16x4_f32(false, a, false, b0, (short)0, acc0, false, false);
        acc1 = __builtin_amdgcn_wmma_f32_16x16x4_f32(false, a, false, b1, (short)0, acc1, false, false);
        acc2 = __builtin_amdgcn_wmma_f32_16x16x4_f32(false, a, false, b2, (short)0, acc2, false, false);
        acc3 = __builtin_amdgcn_wmma_f32_16x16x4_f32(false, a, false, b3, (short)0, acc3, false, false);
    }

    // C layout: VGPR r -> row m0+r (lanes 0-15) / m0+8+r (lanes 16-31)
    int c0 = n0 + lrow;
    int c1 = c0 + 16;
    int c2 = c0 + 32;
    int c3 = c0 + 48;
    bool ok0 = c0 < N, ok1 = c1 < N, ok2 = c2 < N, ok3 = c3 < N;

    #pragma unroll
    for (int r = 0; r < 8; ++r) {
        size_t rowBase = (size_t)(m0 + r + 8 * khalf) * N;
        if (ok0) C[rowBase + c0] = acc0[r];
        if (ok1) C[rowBase + c1] = acc1[r];
        if (ok2) C[rowBase + c2] = acc2[r];
        if (ok3) C[rowBase + c3] = acc3[r];
    }
}

// ---------------- out = h * dinv^2 + b (float4, initializes accumulator) ----------

__global__ void gcn_self_bias(const float* __restrict__ h, const float* __restrict__ dinv,
                              const float* __restrict__ b, float* __restrict__ out,
                              int n, int H4, int H) {
    int t = blockIdx.x * blockDim.x + threadIdx.x;
    if (t >= n * H4) return;
    int row = t / H4;
    int c   = t - row * H4;
    float di = dinv[row];
    float d2 = di * di;
    float4 hv = ((const float4*)(h + (size_t)row * H))[c];
    float4 bv = ((const float4*)b)[c];
    float4 o;
    o.x = hv.x * d2 + bv.x;
    o.y = hv.y * d2 + bv.y;
    o.z = hv.z * d2 + bv.z;
    o.w = hv.w * d2 + bv.w;
    ((float4*)(out + (size_t)row * H))[c] = o;
}

// ---------------- scatter: out[dst] += norm[e] * h[src] ----------------
// One thread per (edge, float4-chunk); H=300 -> 75 chunks, rows are 16B aligned.

__global__ void gcn_scatter(const float* __restrict__ h, const int* __restrict__ src,
                            const int* __restrict__ dst, const float* __restrict__ norm,
                            float* __restrict__ out, int e, int H4, int H) {
    int t = blockIdx.x * blockDim.x + threadIdx.x;
    if (t >= e * H4) return;
    int ed = t / H4;
    int c  = t - ed * H4;
    int s = src[ed];
    int d = dst[ed];
    float w = norm[ed];
    float4 v = ((const float4*)(h + (size_t)s * H))[c];
    float* op = out + (size_t)d * H + 4 * c;
    atomicAdd(op + 0, v.x * w);
    atomicAdd(op + 1, v.y * w);
    atomicAdd(op + 2, v.z * w);
    atomicAdd(op + 3, v.w * w);
}

// ---------------- launch ----------------

extern "C" void kernel_launch(void* const* d_in, const int* in_sizes, int n_in,
                              void* d_out, int out_size, void* d_ws, size_t ws_size,
                              hipStream_t stream) {
    const float* V  = (const float*)d_in[0];
    const int*   ei = (const int*)d_in[1];     // [2, E] : src row then dst row
    const float* W1 = (const float*)d_in[2];
    const float* b1 = (const float*)d_in[3];
    const float* W2 = (const float*)d_in[4];
    const float* b2 = (const float*)d_in[5];
    float* out = (float*)d_out;

    const int* src = ei;
    const int* dst = ei + N_EDGES;

    // workspace layout (floats): dinv | norm | Wp | h | x1   (~124 MB total)
    float* wsf  = (float*)d_ws;
    float* dinv = wsf;                               // 50,000
    float* norm = wsf + 50048;                       // 800,000
    float* Wp   = wsf + 850048;                      // up to 300*320 = 96,000
    float* h    = wsf + 946304;                      // 15,000,000 (h1, reused for h2)
    float* x1   = wsf + 15946304;                    // 15,000,000

    const int TB = 256;
    const int colGroups = NPAD / 64;                 // 5
    const int gemmWaves = (N_NODES / 16) * colGroups;
    const int gemmGrid  = (gemmWaves * 32 + TB - 1) / TB;
    const int H4 = HIDDEN / 4;                       // 75
    const int sbGrid    = (N_NODES * H4 + TB - 1) / TB;
    const int scatGrid  = (N_EDGES * H4 + TB - 1) / TB;

    // degrees & edge norms
    gcn_deg_init <<<(N_NODES + TB - 1) / TB, TB, 0, stream>>>(dinv, N_NODES);
    gcn_deg_accum<<<(N_EDGES + TB - 1) / TB, TB, 0, stream>>>(dinv, dst, N_EDGES);
    gcn_rsqrt    <<<(N_NODES + TB - 1) / TB, TB, 0, stream>>>(dinv, N_NODES);
    gcn_norm     <<<(N_EDGES + TB - 1) / TB, TB, 0, stream>>>(dinv, src, dst, norm, N_EDGES);

    // layer 1
    gcn_pad_w    <<<(N_FEAT * NPAD + TB - 1) / TB, TB, 0, stream>>>(W1, Wp, N_FEAT);
    gcn_gemm_wmma<<<gemmGrid, TB, 0, stream>>>(V, Wp, h, N_NODES, N_FEAT, HIDDEN, colGroups);
    gcn_self_bias<<<sbGrid, TB, 0, stream>>>(h, dinv, b1, x1, N_NODES, H4, HIDDEN);
    gcn_scatter  <<<scatGrid, TB, 0, stream>>>(h, src, dst, norm, x1, N_EDGES, H4, HIDDEN);

    // layer 2
    gcn_pad_w    <<<(HIDDEN * NPAD + TB - 1) / TB, TB, 0, stream>>>(W2, Wp, HIDDEN);
    gcn_gemm_wmma<<<gemmGrid, TB, 0, stream>>>(x1, Wp, h, N_NODES, HIDDEN, HIDDEN, colGroups);
    gcn_self_bias<<<sbGrid, TB, 0, stream>>>(h, dinv, b2, out, N_NODES, H4, HIDDEN);
    gcn_scatter  <<<scatGrid, TB, 0, stream>>>(h, src, dst, norm, out, N_EDGES, H4, HIDDEN);
}